// VQVAE_48885317763940
// MI455X (gfx1250) — compile-verified
//
#include <hip/hip_runtime.h>

// ---------------------------------------------------------------------------
// VQ-VAE forward on gfx1250 (CDNA5). All GEMMs via V_WMMA_F32_16X16X4_F32.
// Whole internal pipeline runs NHWC with pre-packed weights wT[m][tap][ci]:
//  - a WMMA K-step of 4 = 4 consecutive channels -> all A/B fragment loads
//    are contiguous b64 loads.
//  - tap-major K ordering: coords / bounds masks / pointers hoist per tap.
//  - 2x2 register tile per wave (32M x 32N), 4 accumulators.
// Transposed conv = lhs-dilated conv; tap flip + (in,out) transpose folded
// into the weight-pack kernel. e1 (Cin=3) is padded to 4 channels.
// ---------------------------------------------------------------------------

typedef float v2f __attribute__((ext_vector_type(2)));
typedef float v8f __attribute__((ext_vector_type(8)));

__device__ __forceinline__ v8f wmma_f32(v2f a, v2f b, v8f c) {
  return __builtin_amdgcn_wmma_f32_16x16x4_f32(false, a, false, b,
                                               (short)0, c, false, false);
}

// Resolve an input (possibly lhs-dilated) coordinate to a safe element offset
// and a 0/1 validity mask.
template<int LD>
__device__ __forceinline__ void bcoord(int hd, int wd, int Hin, int Win,
                                       int& off, float& mask) {
  bool ok;
  int h, w;
  if (LD == 1) {
    ok = (hd >= 0) && (wd >= 0) && (hd < Hin) && (wd < Win);
    h = ok ? hd : 0;
    w = ok ? wd : 0;
  } else {
    ok = (hd >= 0) && (wd >= 0) && ((hd % LD) == 0) && ((wd % LD) == 0);
    h = hd / LD;
    w = wd / LD;
    ok = ok && (h < Hin) && (w < Win);
    if (!ok) { h = 0; w = 0; }
  }
  off = h * Win + w;
  mask = ok ? 1.0f : 0.0f;
}

// ---------------------------------------------------------------------------
// NHWC implicit-GEMM conv. Requires Cin % 4 == 0, Ncols % 32 == 0, and
// packed weights with M padded to a multiple of 32 (zero rows).
// ONCHW selects NCHW output indexing (only the final layer).
// ---------------------------------------------------------------------------
template<int KH, int KW, int STRIDE, int LD, bool ONCHW>
__global__ __launch_bounds__(256)
void conv_nhwc_wmma(const float* __restrict__ in, const float* __restrict__ wgtT,
                    const float* __restrict__ bias, const float* __restrict__ res,
                    float* __restrict__ out,
                    int Nb, int Cin, int Hin, int Win,
                    int Cout, int Hout, int Wout,
                    int pad, int relu_in, int relu_out)
{
  const int KHW = KH * KW;
  const int K   = Cin * KHW;
  const int HWi = Hin * Win;
  const int HWo = Hout * Wout;
  const int Ncols = Nb * HWo;

  const int wave = threadIdx.x >> 5;
  const int lane = threadIdx.x & 31;
  const int half = lane >> 4;
  const int l16  = lane & 15;

  const int ntiles = Ncols >> 5;
  const int ntile  = blockIdx.x * 8 + wave;
  if (ntile >= ntiles) return;          // uniform per wave
  const int m0   = blockIdx.y * 32;
  const int col0 = ntile * 32;

  // Two B columns per lane.
  const int colA = col0 + l16;
  const int colB = colA + 16;
  const int nA = colA / HWo, sA = colA % HWo, hoA = sA / Wout, woA = sA % Wout;
  const int nB = colB / HWo, sB = colB % HWo, hoB = sB / Wout, woB = sB % Wout;
  const int hbA = hoA * STRIDE - pad, wbA = woA * STRIDE - pad;
  const int hbB = hoB * STRIDE - pad, wbB = woB * STRIDE - pad;
  const float* inA = in + (size_t)nA * HWi * Cin;
  const float* inB = in + (size_t)nB * HWi * Cin;

  // Two A rows per lane (weight rows are zero-padded past Cout).
  const float* wr0base = wgtT + (size_t)(m0 + l16)      * K;
  const float* wr1base = wgtT + (size_t)(m0 + 16 + l16) * K;

  v8f acc00 = {}, acc01 = {}, acc10 = {}, acc11 = {};

  for (int tap = 0; tap < KHW; ++tap) {
    const int kh = tap / KW, kw = tap % KW;

    int offA, offB;
    float mAv, mBv;
    bcoord<LD>(hbA + kh, wbA + kw, Hin, Win, offA, mAv);
    bcoord<LD>(hbB + kh, wbB + kw, Hin, Win, offB, mBv);
    const float* pA = inA + (size_t)offA * Cin;   // channels contiguous
    const float* pB = inB + (size_t)offB * Cin;
    const float* w0 = wr0base + tap * Cin;        // channels contiguous
    const float* w1 = wr1base + tap * Cin;

    for (int cib = 0; cib < Cin; cib += 4) {
      const int c0 = cib + 2 * half;
      v2f a0 = *(const v2f*)(w0 + c0);
      v2f a1 = *(const v2f*)(w1 + c0);
      v2f b0 = *(const v2f*)(pA + c0);
      v2f b1 = *(const v2f*)(pB + c0);
      b0.x *= mAv; b0.y *= mAv;
      b1.x *= mBv; b1.y *= mBv;
      if (relu_in) {
        b0.x = fmaxf(b0.x, 0.0f); b0.y = fmaxf(b0.y, 0.0f);
        b1.x = fmaxf(b1.x, 0.0f); b1.y = fmaxf(b1.y, 0.0f);
      }
      acc00 = wmma_f32(a0, b0, acc00);
      acc01 = wmma_f32(a0, b1, acc01);
      acc10 = wmma_f32(a1, b0, acc10);
      acc11 = wmma_f32(a1, b1, acc11);
    }
  }

  auto store_tile = [&](const v8f& acc, int rbase, int colX, int nX, int hoX, int woX) {
    #pragma unroll
    for (int r = 0; r < 8; ++r) {
      const int row = rbase + r + 8 * half;
      if (row >= Cout) continue;
      float v = acc[r] + bias[row];
      if (relu_out) v = fmaxf(v, 0.0f);
      size_t o;
      if (ONCHW) o = (((size_t)nX * Cout + row) * Hout + hoX) * Wout + woX;
      else       o = (size_t)colX * Cout + row;     // NHWC: contiguous in row
      if (res) v += res[o];
      out[o] = v;
    }
  };
  store_tile(acc00, m0,      colA, nA, hoA, woA);
  store_tile(acc01, m0,      colB, nB, hoB, woB);
  store_tile(acc10, m0 + 16, colA, nA, hoA, woA);
  store_tile(acc11, m0 + 16, colB, nB, hoB, woB);
}

// ---------------------------------------------------------------------------
// Weight packing:  wT[(m*KHW + tap)*CinD + ci]
//   normal conv (OIHW):      = w[((m*CinS + ci)*KHW) + tap]
//   transposed  (I,O,KH,KW): = w[((ci*Cout + m)*KHW) + (KHW-1-tap)]  (flip)
// Rows m in [Cout, Mpad) and channels ci in [CinS, CinD) are zero.
// ---------------------------------------------------------------------------
__global__ void pack_weights(const float* __restrict__ w, float* __restrict__ wT,
                             int Cout, int Mpad, int CinS, int CinD, int KHW,
                             int trans)
{
  const int t = blockIdx.x * blockDim.x + threadIdx.x;
  const int total = Mpad * KHW * CinD;
  if (t >= total) return;
  const int ci  = t % CinD;
  const int tap = (t / CinD) % KHW;
  const int m   = t / (CinD * KHW);
  float v = 0.0f;
  if (m < Cout && ci < CinS) {
    if (trans) v = w[((size_t)ci * Cout + m) * KHW + (KHW - 1 - tap)];
    else       v = w[((size_t)m * CinS + ci) * KHW + tap];
  }
  wT[t] = v;
}

// x (16,3,128,128) NCHW -> NHWC padded to 4 channels (pad channel = 0).
__global__ void pack_x(const float* __restrict__ x, float* __restrict__ xn)
{
  const int t = blockIdx.x * blockDim.x + threadIdx.x;  // 16*16384*4
  if (t >= 16 * 16384 * 4) return;
  const int ci = t & 3;
  const int p  = t >> 2;
  const int n  = p >> 14;
  const int s  = p & 16383;
  xn[t] = (ci < 3) ? x[((size_t)n * 3 + ci) * 16384 + s] : 0.0f;
}

// z_e NHWC -> NCHW (into d_out slice).
__global__ void ze_to_nchw(const float* __restrict__ zn, float* __restrict__ zc)
{
  const int t = blockIdx.x * blockDim.x + threadIdx.x;  // p*256 + c
  if (t >= 16 * 1024 * 256) return;
  const int c = t & 255;
  const int p = t >> 8;
  const int n = p >> 10;
  const int s = p & 1023;
  zc[(((size_t)n * 256 + c) << 10) + s] = zn[t];
}

// ---------------------------------------------------------------------------
// VQ: per-code squared norms.
// ---------------------------------------------------------------------------
__global__ void code_norms(const float* __restrict__ emb, float* __restrict__ norme)
{
  const int j = blockIdx.x * blockDim.x + threadIdx.x;
  if (j >= 512) return;
  const float* e = emb + (size_t)j * 256;
  float sum = 0.0f;
  for (int c = 0; c < 256; ++c) sum += e[c] * e[c];
  norme[j] = sum;
}

// ---------------------------------------------------------------------------
// VQ argmin on NHWC z: scores S[p][j] = ||e_j||^2 - 2 z_p.e_j.
// All fragment loads contiguous b64.
// ---------------------------------------------------------------------------
__global__ __launch_bounds__(256)
void vq_argmin(const float* __restrict__ ze, const float* __restrict__ emb,
               const float* __restrict__ norme, int* __restrict__ idx)
{
  const int wave = threadIdx.x >> 5;
  const int lane = threadIdx.x & 31;
  const int half = lane >> 4;
  const int l16  = lane & 15;

  const int ptile = blockIdx.x * 8 + wave;   // 1024 tiles
  const int p0 = ptile * 16;
  const float* zrow = ze + (size_t)(p0 + l16) * 256;

  float best[8];
  int   bidx[8];
  #pragma unroll
  for (int r = 0; r < 8; ++r) { best[r] = 3.4e38f; bidx[r] = 0; }

  for (int jt = 0; jt < 32; ++jt) {
    const int j0 = jt * 16;
    const float* ecol = emb + (size_t)(j0 + l16) * 256;
    v8f acc = {};
    for (int kb = 0; kb < 256; kb += 4) {
      const int k0 = kb + 2 * half;
      v2f a = *(const v2f*)(zrow + k0);
      v2f b = *(const v2f*)(ecol + k0);
      acc = wmma_f32(a, b, acc);
    }
    const int j = j0 + l16;
    const float nj = norme[j];
    #pragma unroll
    for (int r = 0; r < 8; ++r) {
      const float sc = nj - 2.0f * acc[r];
      if (sc < best[r]) { best[r] = sc; bidx[r] = j; }
    }
  }

  for (int m = 1; m <= 8; m <<= 1) {
    #pragma unroll
    for (int r = 0; r < 8; ++r) {
      const float ov = __shfl_xor(best[r], m, 32);
      const int   oi = __shfl_xor(bidx[r], m, 32);
      if (ov < best[r] || (ov == best[r] && oi < bidx[r])) {
        best[r] = ov; bidx[r] = oi;
      }
    }
  }

  if (l16 == 0) {
    #pragma unroll
    for (int r = 0; r < 8; ++r) idx[p0 + r + 8 * half] = bidx[r];
  }
}

// z_q written in both layouts: NHWC (decoder input) and NCHW (d_out slice).
__global__ void vq_gather2(const int* __restrict__ idx, const float* __restrict__ emb,
                           float* __restrict__ zq_nhwc, float* __restrict__ zq_nchw)
{
  const int t = blockIdx.x * blockDim.x + threadIdx.x;  // p*256 + c
  if (t >= 16 * 1024 * 256) return;
  const int c = t & 255;
  const int p = t >> 8;
  const float v = emb[(size_t)idx[p] * 256 + c];
  zq_nhwc[t] = v;
  const int n = p >> 10;
  const int s = p & 1023;
  zq_nchw[(((size_t)n * 256 + c) << 10) + s] = v;
}

// ---------------------------------------------------------------------------
// Host-side orchestration.
// ---------------------------------------------------------------------------
extern "C" void kernel_launch(void* const* d_in, const int* in_sizes, int n_in,
                              void* d_out, int out_size, void* d_ws, size_t ws_size,
                              hipStream_t stream)
{
  (void)in_sizes; (void)n_in; (void)out_size; (void)ws_size;

  const float* x      = (const float*)d_in[0];
  const float* e1_w   = (const float*)d_in[1];
  const float* e1_b   = (const float*)d_in[2];
  const float* e2_w   = (const float*)d_in[3];
  const float* e2_b   = (const float*)d_in[4];
  const float* e3_w   = (const float*)d_in[5];
  const float* e3_b   = (const float*)d_in[6];
  const float* er1_w1 = (const float*)d_in[7];
  const float* er1_b1 = (const float*)d_in[8];
  const float* er1_w2 = (const float*)d_in[9];
  const float* er1_b2 = (const float*)d_in[10];
  const float* er2_w1 = (const float*)d_in[11];
  const float* er2_b1 = (const float*)d_in[12];
  const float* er2_w2 = (const float*)d_in[13];
  const float* er2_b2 = (const float*)d_in[14];
  const float* dr1_w1 = (const float*)d_in[15];
  const float* dr1_b1 = (const float*)d_in[16];
  const float* dr1_w2 = (const float*)d_in[17];
  const float* dr1_b2 = (const float*)d_in[18];
  const float* dr2_w1 = (const float*)d_in[19];
  const float* dr2_b1 = (const float*)d_in[20];
  const float* dr2_w2 = (const float*)d_in[21];
  const float* dr2_b2 = (const float*)d_in[22];
  const float* emb    = (const float*)d_in[23];
  const float* d1_w   = (const float*)d_in[24];
  const float* d1_b   = (const float*)d_in[25];
  const float* dt1_w  = (const float*)d_in[26];
  const float* dt1_b  = (const float*)d_in[27];
  const float* dt2_w  = (const float*)d_in[28];
  const float* dt2_b  = (const float*)d_in[29];

  // Output slices: (x_hat, z_e, z_q) concatenated flat (all NCHW).
  float* x_hat = (float*)d_out;                        // 16*3*128*128
  float* z_e   = x_hat + (size_t)16 * 3 * 128 * 128;   // 16*256*32*32
  float* z_q   = z_e + (size_t)16 * 256 * 32 * 32;

  // Workspace carve-up (~180 MB of floats).
  float* W0  = (float*)d_ws;                        // NHWC 64x64x256 (16.8M)
  float* W1  = W0  + (size_t)16 * 4096 * 256;
  float* W2  = W1  + (size_t)16 * 1024 * 256;
  float* W3  = W2  + (size_t)16 * 1024 * 256;
  float* Wze = W3  + (size_t)16 * 1024 * 256;       // z_e NHWC
  float* Wzq = Wze + (size_t)16 * 1024 * 256;       // z_q NHWC
  float* Xn  = Wzq + (size_t)16 * 1024 * 256;       // x NHWC, C padded to 4
  float* norme = Xn + (size_t)16 * 16384 * 4;       // 512
  int*   idxb  = (int*)(norme + 512);               // 16384
  float* wTp   = (float*)(idxb + 16384);
  float* e1T   = wTp;                 // 256*16*4     = 16384
  float* e2T   = e1T   + 16384;       // 256*16*256   = 1048576
  float* e3T   = e2T   + 1048576;     // 256*9*256    = 589824
  float* er1aT = e3T   + 589824;
  float* er1bT = er1aT + 589824;      // 256*1*256    = 65536
  float* er2aT = er1bT + 65536;
  float* er2bT = er2aT + 589824;
  float* d1T   = er2bT + 65536;
  float* dr1aT = d1T   + 589824;
  float* dr1bT = dr1aT + 589824;
  float* dr2aT = dr1bT + 65536;
  float* dr2bT = dr2aT + 589824;
  float* dt1T  = dr2bT + 65536;       // 256*16*256   = 1048576
  float* dt2T  = dt1T  + 1048576;     // 32*16*256    = 131072

  const dim3 blk(256);

  // ---- pack weights + input (independent; head of the graph) ----
  pack_weights<<<dim3(64),   blk, 0, stream>>>(e1_w,   e1T,   256, 256, 3,   4,   16, 0);
  pack_weights<<<dim3(4096), blk, 0, stream>>>(e2_w,   e2T,   256, 256, 256, 256, 16, 0);
  pack_weights<<<dim3(2304), blk, 0, stream>>>(e3_w,   e3T,   256, 256, 256, 256, 9,  0);
  pack_weights<<<dim3(2304), blk, 0, stream>>>(er1_w1, er1aT, 256, 256, 256, 256, 9,  0);
  pack_weights<<<dim3(256),  blk, 0, stream>>>(er1_w2, er1bT, 256, 256, 256, 256, 1,  0);
  pack_weights<<<dim3(2304), blk, 0, stream>>>(er2_w1, er2aT, 256, 256, 256, 256, 9,  0);
  pack_weights<<<dim3(256),  blk, 0, stream>>>(er2_w2, er2bT, 256, 256, 256, 256, 1,  0);
  pack_weights<<<dim3(2304), blk, 0, stream>>>(d1_w,   d1T,   256, 256, 256, 256, 9,  0);
  pack_weights<<<dim3(2304), blk, 0, stream>>>(dr1_w1, dr1aT, 256, 256, 256, 256, 9,  0);
  pack_weights<<<dim3(256),  blk, 0, stream>>>(dr1_w2, dr1bT, 256, 256, 256, 256, 1,  0);
  pack_weights<<<dim3(2304), blk, 0, stream>>>(dr2_w1, dr2aT, 256, 256, 256, 256, 9,  0);
  pack_weights<<<dim3(256),  blk, 0, stream>>>(dr2_w2, dr2bT, 256, 256, 256, 256, 1,  0);
  pack_weights<<<dim3(4096), blk, 0, stream>>>(dt1_w,  dt1T,  256, 256, 256, 256, 16, 1);
  pack_weights<<<dim3(512),  blk, 0, stream>>>(dt2_w,  dt2T,  3,   32,  256, 256, 16, 1);
  pack_x<<<dim3(4096), blk, 0, stream>>>(x, Xn);
  code_norms<<<dim3(2), blk, 0, stream>>>(emb, norme);

  // Fast-kernel grids: x = (Ncols/32)/8 wave-tiles, y = Mpad/32 row tiles.
  const dim3 f32x32(64, 8);     // Ncols=16384 (32x32 out), Cout=256
  const dim3 f64x64(256, 8);    // Ncols=65536 (64x64 out), Cout=256
  const dim3 f128(1024, 1);     // Ncols=262144 (128x128 out), Cout=3 (Mpad 32)

  // ---- encoder (NHWC) ----
  conv_nhwc_wmma<4,4,2,1,false><<<f64x64, blk, 0, stream>>>(Xn, e1T, e1_b, nullptr, W0,
      16, 4, 128, 128, 256, 64, 64, 1, 0, 1);
  conv_nhwc_wmma<4,4,2,1,false><<<f32x32, blk, 0, stream>>>(W0, e2T, e2_b, nullptr, W1,
      16, 256, 64, 64, 256, 32, 32, 1, 0, 1);
  conv_nhwc_wmma<3,3,1,1,false><<<f32x32, blk, 0, stream>>>(W1, e3T, e3_b, nullptr, W2,
      16, 256, 32, 32, 256, 32, 32, 1, 0, 0);
  // er1
  conv_nhwc_wmma<3,3,1,1,false><<<f32x32, blk, 0, stream>>>(W2, er1aT, er1_b1, nullptr, W3,
      16, 256, 32, 32, 256, 32, 32, 1, 1, 1);
  conv_nhwc_wmma<1,1,1,1,false><<<f32x32, blk, 0, stream>>>(W3, er1bT, er1_b2, W2, W1,
      16, 256, 32, 32, 256, 32, 32, 0, 0, 0);
  // er2 -> z_e (NHWC), then layout copy into d_out
  conv_nhwc_wmma<3,3,1,1,false><<<f32x32, blk, 0, stream>>>(W1, er2aT, er2_b1, nullptr, W3,
      16, 256, 32, 32, 256, 32, 32, 1, 1, 1);
  conv_nhwc_wmma<1,1,1,1,false><<<f32x32, blk, 0, stream>>>(W3, er2bT, er2_b2, W1, Wze,
      16, 256, 32, 32, 256, 32, 32, 0, 0, 0);
  ze_to_nchw<<<dim3(16384), blk, 0, stream>>>(Wze, z_e);

  // ---- VQ ----
  vq_argmin<<<dim3(128), blk, 0, stream>>>(Wze, emb, norme, idxb);
  vq_gather2<<<dim3(16384), blk, 0, stream>>>(idxb, emb, Wzq, z_q);

  // ---- decoder (forward dec_in == z_q) ----
  conv_nhwc_wmma<3,3,1,1,false><<<f32x32, blk, 0, stream>>>(Wzq, d1T, d1_b, nullptr, W1,
      16, 256, 32, 32, 256, 32, 32, 1, 0, 0);
  // dr1
  conv_nhwc_wmma<3,3,1,1,false><<<f32x32, blk, 0, stream>>>(W1, dr1aT, dr1_b1, nullptr, W2,
      16, 256, 32, 32, 256, 32, 32, 1, 1, 1);
  conv_nhwc_wmma<1,1,1,1,false><<<f32x32, blk, 0, stream>>>(W2, dr1bT, dr1_b2, W1, W3,
      16, 256, 32, 32, 256, 32, 32, 0, 0, 0);
  // dr2
  conv_nhwc_wmma<3,3,1,1,false><<<f32x32, blk, 0, stream>>>(W3, dr2aT, dr2_b1, nullptr, W2,
      16, 256, 32, 32, 256, 32, 32, 1, 1, 1);
  conv_nhwc_wmma<1,1,1,1,false><<<f32x32, blk, 0, stream>>>(W2, dr2bT, dr2_b2, W3, W1,
      16, 256, 32, 32, 256, 32, 32, 0, 0, 0);
  // dt1: transposed conv 4x4 s2 p1 -> lhs_dilation=2, pad' = 2, relu
  conv_nhwc_wmma<4,4,1,2,false><<<f64x64, blk, 0, stream>>>(W1, dt1T, dt1_b, nullptr, W0,
      16, 256, 32, 32, 256, 64, 64, 2, 0, 1);
  // dt2: transposed conv 4x4 s2 p1 -> x_hat (NCHW, Cout=3)
  conv_nhwc_wmma<4,4,1,2,true><<<f128, blk, 0, stream>>>(W0, dt2T, dt2_b, nullptr, x_hat,
      16, 256, 64, 64, 3, 128, 128, 2, 0, 0);
}